// G2_24601572672050
// MI455X (gfx1250) — compile-verified
//
#include <hip/hip_runtime.h>
#include <stdint.h>
#include <math.h>

typedef __attribute__((ext_vector_type(2))) float v2f;
typedef __attribute__((ext_vector_type(8))) float v8f;

#define DD 64

// ---------------- zero ----------------
__global__ void G2_zero_f32(float* __restrict__ p, long long n) {
    long long i = (long long)blockIdx.x * blockDim.x + threadIdx.x;
    if (i < n) p[i] = 0.0f;
}

// ---------------- phase 1: scatter X[src] -> sum[dst], cnt[dst] ----------------
__global__ void G2_scatter_x(const float* __restrict__ X,
                             const long long* __restrict__ ei,
                             float* __restrict__ sum, float* __restrict__ cnt,
                             long long E) {
    long long g = (long long)blockIdx.x * blockDim.x + threadIdx.x;
    long long e = g >> 6;
    int d = (int)(g & 63);
    if (e >= E) return;
    int s = (int)ei[e];        // src
    int t = (int)ei[E + e];    // dst
    atomicAdd(&sum[(long long)t * DD + d], X[(long long)s * DD + d]);
    if (d == 0) atomicAdd(&cnt[t], 1.0f);
}

// ---------------- phase 2: h = relu([agg|X] @ [Wn;Ws] + b) via V_WMMA_F32_16X16X4_F32 ----
__global__ __launch_bounds__(128)
void G2_sage_h_wmma(const float* __restrict__ X,
                    const float* __restrict__ aggsum,
                    const float* __restrict__ cnt,
                    const float* __restrict__ Wn,
                    const float* __restrict__ Ws,
                    const float* __restrict__ bias,
                    float* __restrict__ h,
                    int nTiles, int N) {
    __shared__ float ldsW[128 * DD];   // fused weight matrix [Wn ; Ws], 32 KB
    __shared__ float ldsB[DD];

    int tid = threadIdx.x;
    // stage weights into LDS with float4 (2048 float4 / 128 threads = 16 iters)
    {
        const float4* wn4 = (const float4*)Wn;
        const float4* ws4 = (const float4*)Ws;
        float4* l4 = (float4*)ldsW;
        for (int i = tid; i < 2048; i += 128)
            l4[i] = (i < 1024) ? wn4[i] : ws4[i - 1024];
        if (tid < DD) ldsB[tid] = bias[tid];
    }
    __syncthreads();

    int wave = tid >> 5;
    int lane = tid & 31;
    int tile = blockIdx.x * 4 + wave;
    if (tile >= nTiles) return;

    int half = lane >> 4;        // 0: K pair {0,1}; 1: K pair {2,3}
    int ln   = lane & 15;        // A: row-in-tile, B/C/D: column
    int kSel = half * 2;
    int row  = tile * 16 + ln;
    int rowc = row < N ? row : N - 1;  // clamp for safety (N is a multiple of 16 here)

    float invc = 1.0f / fmaxf(cnt[rowc], 1.0f);
    const float* arow = aggsum + (long long)rowc * DD;
    const float* xrow = X      + (long long)rowc * DD;

    v8f acc[4] = {};

    // K = 0..63 : aggregated-neighbor half (scaled by 1/count -> mean)
    #pragma unroll
    for (int kk = 0; kk < 16; ++kk) {
        int kb = kk * 4 + kSel;
        v2f a; a.x = arow[kb] * invc; a.y = arow[kb + 1] * invc;
        const float* w0 = ldsW + kb * DD;
        const float* w1 = ldsW + (kb + 1) * DD;
        #pragma unroll
        for (int j = 0; j < 4; ++j) {
            v2f b; b.x = w0[j * 16 + ln]; b.y = w1[j * 16 + ln];
            acc[j] = __builtin_amdgcn_wmma_f32_16x16x4_f32(
                false, a, false, b, (short)0, acc[j], false, false);
        }
    }
    // K = 64..127 : self half (X @ W_self)
    #pragma unroll
    for (int kk = 16; kk < 32; ++kk) {
        int kb = kk * 4 + kSel;
        v2f a; a.x = xrow[kb - 64]; a.y = xrow[kb - 63];
        const float* w0 = ldsW + kb * DD;
        const float* w1 = ldsW + (kb + 1) * DD;
        #pragma unroll
        for (int j = 0; j < 4; ++j) {
            v2f b; b.x = w0[j * 16 + ln]; b.y = w1[j * 16 + ln];
            acc[j] = __builtin_amdgcn_wmma_f32_16x16x4_f32(
                false, a, false, b, (short)0, acc[j], false, false);
        }
    }

    // epilogue: C/D layout lane->N=ln, VGPR v -> M = half*8 + v
    #pragma unroll
    for (int j = 0; j < 4; ++j) {
        float bb = ldsB[j * 16 + ln];
        #pragma unroll
        for (int v = 0; v < 8; ++v) {
            int m = tile * 16 + half * 8 + v;
            if (m < N) {
                float val = fmaxf(acc[j][v] + bb, 0.0f);
                h[(long long)m * DD + j * 16 + ln] = val;
            }
        }
    }
}

// ---------------- phase 3: m = (h[src]-h[dst])^2 scatter-> sum[src], cnt[src] -----
__global__ void G2_scatter_diff(const float* __restrict__ h,
                                const long long* __restrict__ ei,
                                float* __restrict__ sum, float* __restrict__ cnt,
                                long long E) {
    long long g = (long long)blockIdx.x * blockDim.x + threadIdx.x;
    long long e = g >> 6;
    int d = (int)(g & 63);
    if (e >= E) return;
    int s = (int)ei[e];
    int t = (int)ei[E + e];
    float df = h[(long long)s * DD + d] - h[(long long)t * DD + d];
    float m = df * df;   // P == 2.0
    atomicAdd(&sum[(long long)s * DD + d], m);
    if (d == 0) atomicAdd(&cnt[s], 1.0f);
}

// ---------------- phase 4: out = tanh(sum / max(cnt,1)) ----------------
__global__ void G2_finalize(const float* __restrict__ sum,
                            const float* __restrict__ cnt,
                            float* __restrict__ out, long long n64) {
    long long i = (long long)blockIdx.x * blockDim.x + threadIdx.x;
    if (i >= n64) return;
    float c = fmaxf(cnt[i >> 6], 1.0f);
    out[i] = tanhf(sum[i] / c);
}

extern "C" void kernel_launch(void* const* d_in, const int* in_sizes, int n_in,
                              void* d_out, int out_size, void* d_ws, size_t ws_size,
                              hipStream_t stream) {
    const float*     X  = (const float*)d_in[0];
    const long long* ei = (const long long*)d_in[1];
    const float*     Wn = (const float*)d_in[2];
    const float*     Ws = (const float*)d_in[3];
    const float*     b  = (const float*)d_in[4];
    float* out = (float*)d_out;

    const int       N = in_sizes[0] / DD;      // 50000
    const long long E = in_sizes[1] / 2;       // 800000

    float* ws     = (float*)d_ws;
    float* aggsum = ws;                          // N*64   (reused as msum)
    float* h      = ws + (long long)N * DD;      // N*64
    float* cnt    = ws + (long long)2 * N * DD;  // N

    const long long nd  = (long long)N * DD;
    const int  B256     = 256;
    const unsigned gZnd = (unsigned)((nd + B256 - 1) / B256);
    const unsigned gZn  = (unsigned)((N + B256 - 1) / B256);
    const long long eth = E * DD;
    const unsigned gE   = (unsigned)((eth + B256 - 1) / B256);
    const int nTiles    = (N + 15) / 16;
    const unsigned gW   = (unsigned)((nTiles + 3) / 4);

    // phase 1: neighbor mean accumulation
    G2_zero_f32<<<gZnd, B256, 0, stream>>>(aggsum, nd);
    G2_zero_f32<<<gZn,  B256, 0, stream>>>(cnt, N);
    G2_scatter_x<<<gE, B256, 0, stream>>>(X, ei, aggsum, cnt, E);

    // phase 2: fused dual-GEMM + bias + relu via WMMA f32
    G2_sage_h_wmma<<<gW, 128, 0, stream>>>(X, aggsum, cnt, Wn, Ws, b, h, nTiles, N);

    // phase 3: edge |diff|^2 scatter-mean to src
    G2_zero_f32<<<gZnd, B256, 0, stream>>>(aggsum, nd);
    G2_zero_f32<<<gZn,  B256, 0, stream>>>(cnt, N);
    G2_scatter_diff<<<gE, B256, 0, stream>>>(h, ei, aggsum, cnt, E);

    // phase 4: tanh(mean)
    G2_finalize<<<gZnd, B256, 0, stream>>>(aggsum, cnt, out, nd);
}